// WARPLoss_28432683500213
// MI455X (gfx1250) — compile-verified
//
#include <hip/hip_runtime.h>
#include <stdint.h>

#define TMAX 50
#define MARGIN_F 1.0f

// ---------------------------------------------------------------------------
// JAX threefry2x32 with key = PRNGKey(42) = (0, 42), 20 rounds (5x4).
// Partitionable mode (modern JAX default): bits(f) = o0 ^ o1 of
// threefry2x32((0,42), (hi=0, lo=f)) for flat index f < 2^32.
// ---------------------------------------------------------------------------
#define TF_ROUND(r) { x0 += x1; x1 = (x1 << (r)) | (x1 >> (32 - (r))); x1 ^= x0; }

__device__ __forceinline__ uint32_t jax_bits_0_42(uint32_t f) {
    const uint32_t ks0 = 0u;
    const uint32_t ks1 = 42u;
    const uint32_t ks2 = 0u ^ 42u ^ 0x1BD11BDAu;
    uint32_t x0 = 0u + ks0;     // counts_hi
    uint32_t x1 = f + ks1;      // counts_lo
    TF_ROUND(13) TF_ROUND(15) TF_ROUND(26) TF_ROUND(6)
    x0 += ks1; x1 += ks2 + 1u;
    TF_ROUND(17) TF_ROUND(29) TF_ROUND(16) TF_ROUND(24)
    x0 += ks2; x1 += ks0 + 2u;
    TF_ROUND(13) TF_ROUND(15) TF_ROUND(26) TF_ROUND(6)
    x0 += ks0; x1 += ks1 + 3u;
    TF_ROUND(17) TF_ROUND(29) TF_ROUND(16) TF_ROUND(24)
    x0 += ks1; x1 += ks2 + 4u;
    TF_ROUND(13) TF_ROUND(15) TF_ROUND(26) TF_ROUND(6)
    x0 += ks2; x1 += ks0 + 5u;
    return x0 ^ x1;
}

// ---------------------------------------------------------------------------
// Kernel 1: per-block (1024-element chunk) count of label==0.
// Uses gfx1250 async global->LDS DMA (ASYNCcnt path) to stage label tiles.
// ---------------------------------------------------------------------------
__global__ __launch_bounds__(256) void k_count_zeros(const int* __restrict__ labels,
                                                     unsigned* __restrict__ blockCounts,
                                                     int n) {
    __shared__ int lds_lab[256 * 4];
    __shared__ unsigned red[256];
    const int t  = (int)threadIdx.x;
    const int i0 = (int)blockIdx.x * 1024 + t * 4;
    unsigned c = 0u;
    if (i0 + 3 < n) {
        // CDNA5 async copy: 16B of labels per lane, global -> LDS, then wait
        // on the wave's ASYNCcnt (split-counter model, gfx1250).
        unsigned lds_off = (unsigned)(uintptr_t)(&lds_lab[t * 4]); // low 32b of flat = LDS offset
        unsigned long long gaddr = (unsigned long long)(uintptr_t)(labels + i0);
        asm volatile(
            "global_load_async_to_lds_b128 %0, %1, off\n\t"
            "s_wait_asynccnt 0x0"
            :
            : "v"(lds_off), "v"(gaddr)
            : "memory");
        int4 lab4 = *reinterpret_cast<const int4*>(&lds_lab[t * 4]);
        c = (unsigned)(lab4.x == 0) + (unsigned)(lab4.y == 0) +
            (unsigned)(lab4.z == 0) + (unsigned)(lab4.w == 0);
    } else {
        for (int k = 0; k < 4; ++k) {
            int i = i0 + k;
            if (i < n) c += (unsigned)(labels[i] == 0);
        }
    }
    red[t] = c;
    __syncthreads();
    for (int s = 128; s > 0; s >>= 1) {
        if (t < s) red[t] += red[t + s];
        __syncthreads();
    }
    if (t == 0) blockCounts[blockIdx.x] = red[0];
}

// ---------------------------------------------------------------------------
// Kernel 2: single-block exclusive scan over block counts; emits num_neg.
// ---------------------------------------------------------------------------
__global__ __launch_bounds__(256) void k_scan_blocks(const unsigned* __restrict__ blockCounts,
                                                     unsigned* __restrict__ blockOffsets,
                                                     int* __restrict__ p_num_neg,
                                                     int nb) {
    __shared__ unsigned red[256];
    const int t = (int)threadIdx.x;
    const int K = (nb + 255) / 256;
    unsigned c = 0u;
    for (int k = 0; k < K; ++k) {
        int b = t * K + k;
        if (b < nb) c += blockCounts[b];
    }
    red[t] = c;
    __syncthreads();
    // Hillis-Steele inclusive scan (read-old / barrier / add / barrier)
    for (int s = 1; s < 256; s <<= 1) {
        unsigned v = (t >= s) ? red[t - s] : 0u;
        __syncthreads();
        red[t] += v;
        __syncthreads();
    }
    unsigned run = red[t] - c;  // exclusive
    for (int k = 0; k < K; ++k) {
        int b = t * K + k;
        if (b < nb) {
            blockOffsets[b] = run;
            run += blockCounts[b];
        }
    }
    if (t == 255) *p_num_neg = (int)red[255];
}

// ---------------------------------------------------------------------------
// Kernel 3: stable compaction — neg_scores[rank] = scores[i] for label[i]==0,
// rank in ascending index order (matches stable argsort(labels): zeros first).
// ---------------------------------------------------------------------------
__global__ __launch_bounds__(256) void k_scatter_negs(const float* __restrict__ scores,
                                                      const int* __restrict__ labels,
                                                      const unsigned* __restrict__ blockOffsets,
                                                      float* __restrict__ neg_scores,
                                                      int n) {
    __shared__ unsigned red[256];
    const int t  = (int)threadIdx.x;
    const int i0 = (int)blockIdx.x * 1024 + t * 4;
    int   lab[4];
    float sc[4];
    unsigned c = 0u;
    if (i0 + 3 < n) {
        int4   l4 = *reinterpret_cast<const int4*>(labels + i0);
        float4 s4 = *reinterpret_cast<const float4*>(scores + i0);
        lab[0] = l4.x; lab[1] = l4.y; lab[2] = l4.z; lab[3] = l4.w;
        sc[0]  = s4.x; sc[1]  = s4.y; sc[2]  = s4.z; sc[3]  = s4.w;
    } else {
        for (int k = 0; k < 4; ++k) {
            int i = i0 + k;
            lab[k] = (i < n) ? labels[i] : 1;
            sc[k]  = (i < n) ? scores[i] : 0.0f;
        }
    }
    for (int k = 0; k < 4; ++k) c += (unsigned)(lab[k] == 0);

    red[t] = c;
    __syncthreads();
    for (int s = 1; s < 256; s <<= 1) {
        unsigned v = (t >= s) ? red[t - s] : 0u;
        __syncthreads();
        red[t] += v;
        __syncthreads();
    }
    unsigned base = blockOffsets[blockIdx.x] + (red[t] - c);
    for (int k = 0; k < 4; ++k) {
        if (lab[k] == 0) neg_scores[base++] = sc[k];
    }
}

// ---------------------------------------------------------------------------
// Kernel 4: main WARP loop. One thread per example; rejection-sample with
// early exit at the first margin violation (expected ~1-2 trials, vs the
// reference's dense B x 50 materialization). Per-block partial sum.
// ---------------------------------------------------------------------------
__global__ __launch_bounds__(256) void k_warp_main(const float* __restrict__ scores,
                                                   const int* __restrict__ labels,
                                                   const float* __restrict__ neg_scores,
                                                   const int* __restrict__ p_num_neg,
                                                   float* __restrict__ partials,
                                                   int n) {
    __shared__ float red[256];
    const int i = (int)blockIdx.x * 256 + (int)threadIdx.x;
    const int num_neg = *p_num_neg;
    float per = 0.0f;
    if (i < n && num_neg > 0 && labels[i] == 1) {
        const float sc   = scores[i];
        const float negf = (float)num_neg;
        const uint32_t fbase = (uint32_t)i * (uint32_t)TMAX;
        for (int t = 0; t < TMAX; ++t) {
            uint32_t bits = jax_bits_0_42(fbase + (uint32_t)t);
            float u = __uint_as_float((bits >> 9) | 0x3f800000u) - 1.0f;
            int idx = (int)(u * negf);
            if (idx > num_neg - 1) idx = num_neg - 1;
            float neg = neg_scores[idx];
            if (neg + MARGIN_F > sc) {
                int rank = TMAX / (t + 1);
                if (rank < 1) rank = 1;
                float h = 0.0f;
                for (int k = 1; k <= rank; ++k) h += 1.0f / (float)k;
                per = h * fmaxf(MARGIN_F - (sc - neg), 0.0f);
                break;
            }
        }
    }
    red[threadIdx.x] = per;
    __syncthreads();
    for (int s = 128; s > 0; s >>= 1) {
        if ((int)threadIdx.x < s) red[threadIdx.x] += red[threadIdx.x + s];
        __syncthreads();
    }
    if (threadIdx.x == 0) partials[blockIdx.x] = red[0];
}

// ---------------------------------------------------------------------------
// Kernel 5: deterministic final reduction + normalization.
// ---------------------------------------------------------------------------
__global__ __launch_bounds__(256) void k_finalize(const float* __restrict__ partials,
                                                  int nparts,
                                                  const int* __restrict__ p_num_neg,
                                                  int n,
                                                  float* __restrict__ out) {
    __shared__ float red[256];
    const int t = (int)threadIdx.x;
    float s = 0.0f;
    for (int p = t; p < nparts; p += 256) s += partials[p];
    red[t] = s;
    __syncthreads();
    for (int k = 128; k > 0; k >>= 1) {
        if (t < k) red[t] += red[t + k];
        __syncthreads();
    }
    if (t == 0) {
        int num_neg = *p_num_neg;
        int num_pos = n - num_neg;
        float r = 0.0f;
        if (num_pos > 0 && num_neg > 0) r = red[0] / (float)num_pos;
        out[0] = r;
    }
}

// ---------------------------------------------------------------------------
extern "C" void kernel_launch(void* const* d_in, const int* in_sizes, int n_in,
                              void* d_out, int out_size, void* d_ws, size_t ws_size,
                              hipStream_t stream) {
    (void)n_in; (void)out_size; (void)ws_size;
    const float* scores = (const float*)d_in[0];
    const int*   labels = (const int*)d_in[1];
    const int n = in_sizes[0];

    const int nb1 = (n + 1023) / 1024;   // 1024-element chunks
    const int nb4 = (n + 255) / 256;     // main-kernel blocks

    // Workspace layout (all deterministic, fully written before read):
    char* ws = (char*)d_ws;
    unsigned* blockCounts  = (unsigned*)ws;
    unsigned* blockOffsets = blockCounts + nb1;
    int*      p_num_neg    = (int*)(blockOffsets + nb1);
    size_t hdr   = (size_t)(2 * nb1 + 1) * sizeof(unsigned);
    size_t negOf = (hdr + 255) & ~(size_t)255;
    float* neg_scores = (float*)(ws + negOf);
    size_t partOf = (negOf + (size_t)n * sizeof(float) + 255) & ~(size_t)255;
    float* partials = (float*)(ws + partOf);

    k_count_zeros <<<nb1, 256, 0, stream>>>(labels, blockCounts, n);
    k_scan_blocks <<<1,   256, 0, stream>>>(blockCounts, blockOffsets, p_num_neg, nb1);
    k_scatter_negs<<<nb1, 256, 0, stream>>>(scores, labels, blockOffsets, neg_scores, n);
    k_warp_main   <<<nb4, 256, 0, stream>>>(scores, labels, neg_scores, p_num_neg, partials, n);
    k_finalize    <<<1,   256, 0, stream>>>(partials, nb4, p_num_neg, n, (float*)d_out);
}